// GridNeRFRenderer_17514876634250
// MI455X (gfx1250) — compile-verified
//
#include <hip/hip_runtime.h>
#include <stdint.h>

// Problem constants (fixed by the reference harness).
#define SC   128     // coarse samples per ray
#define NFS  64      // fine samples per ray
#define WAVES_PER_BLOCK 8
#define THREADS (WAVES_PER_BLOCK * 32)

// ---- CDNA5 helpers -------------------------------------------------------

// Byte offset of a generic pointer inside LDS (AS3 addresses are offsets
// from the wave's LDS base, which is what GLOBAL_LOAD_ASYNC_TO_LDS wants).
__device__ __forceinline__ uint32_t lds_off(const void* p) {
  return (uint32_t)(uintptr_t)(__attribute__((address_space(3))) const void*)p;
}

// Async 16B global -> LDS copy per lane (VGLOBAL GLOBAL_LOAD_ASYNC_TO_LDS_B128,
// GV mode: vdst = per-lane LDS byte address, vaddr = per-lane 64-bit address).
// Tracked with ASYNCcnt.
__device__ __forceinline__ void async_copy16(uint32_t lds_byte_addr, const void* gptr) {
  uint64_t ga = (uint64_t)(uintptr_t)gptr;
  asm volatile("global_load_async_to_lds_b128 %0, %1, off"
               :: "v"(lds_byte_addr), "v"(ga)
               : "memory");
}

__device__ __forceinline__ void wait_async_le2() {
  asm volatile("s_wait_asynccnt 2" ::: "memory");
}
__device__ __forceinline__ void wait_async_0() {
  asm volatile("s_wait_asynccnt 0" ::: "memory");
}
__device__ __forceinline__ void wait_lds_0() {
  asm volatile("s_wait_dscnt 0" ::: "memory");
}

// ---- Kernel --------------------------------------------------------------

__global__ __launch_bounds__(THREADS)
void nerf_fine_sample_kernel(const float* __restrict__ rayo,
                             const float* __restrict__ rayd,
                             const float* __restrict__ cw,
                             const float* __restrict__ ct,
                             float* __restrict__ outp,   // [N, NFS, 3]
                             float* __restrict__ outt,   // [N, NFS]
                             int n) {
  // Per-wave double-buffered staging for the streamed rows + CDF scratch.
  __shared__ float s_w[WAVES_PER_BLOCK][2][SC];
  __shared__ float s_t[WAVES_PER_BLOCK][2][SC];
  __shared__ float s_cdf[WAVES_PER_BLOCK][SC + 4];   // 129 used, padded

  const int lane = threadIdx.x & 31;
  const int wv   = threadIdx.x >> 5;
  const int waveGlobal = blockIdx.x * WAVES_PER_BLOCK + wv;
  const int totalWaves = gridDim.x * WAVES_PER_BLOCK;

  const int ray0 = waveGlobal;
  if (ray0 >= n) return;   // wave-uniform

  // Prologue: kick off async stream of ray0's weight row and t row.
  async_copy16(lds_off(&s_w[wv][0][lane * 4]), cw + (size_t)ray0 * SC + lane * 4);
  async_copy16(lds_off(&s_t[wv][0][lane * 4]), ct + (size_t)ray0 * SC + lane * 4);

  int buf = 0;
  for (int ray = ray0; ray < n; ray += totalWaves, buf ^= 1) {
    const int nray = ray + totalWaves;
    if (nray < n) {
      // Software pipeline: next ray's rows go to the other buffer while we
      // compute on this one. Also prefetch next ray's o/d cachelines.
      async_copy16(lds_off(&s_w[wv][buf ^ 1][lane * 4]), cw + (size_t)nray * SC + lane * 4);
      async_copy16(lds_off(&s_t[wv][buf ^ 1][lane * 4]), ct + (size_t)nray * SC + lane * 4);
      __builtin_prefetch(rayo + (size_t)nray * 3, 0, 0);
      __builtin_prefetch(rayd + (size_t)nray * 3, 0, 0);
      wait_async_le2();   // ray's 2 copies done; next ray's 2 still in flight
    } else {
      wait_async_0();
    }

    // ---- per-ray CDF: lane holds 4 weights, wave32 shuffle scan ----
    const float* wrow = &s_w[wv][buf][0];
    float w0 = wrow[lane * 4 + 0] + 1e-5f;
    float w1 = wrow[lane * 4 + 1] + 1e-5f;
    float w2 = wrow[lane * 4 + 2] + 1e-5f;
    float w3 = wrow[lane * 4 + 3] + 1e-5f;
    float p0 = w0;
    float p1 = p0 + w1;
    float p2 = p1 + w2;
    float p3 = p2 + w3;          // lane-local inclusive partials
    float inc = p3;              // inclusive scan of lane sums
#pragma unroll
    for (int d = 1; d < 32; d <<= 1) {
      float o = __shfl_up(inc, (unsigned)d, 32);
      if (lane >= d) inc += o;
    }
    const float total = __shfl(inc, 31, 32);
    const float excl  = inc - p3;
    const float invT  = 1.0f / total;

    // cdf[0] = 0, cdf[1..128] = normalized cumsum
    if (lane == 0) s_cdf[wv][0] = 0.0f;
    s_cdf[wv][1 + lane * 4 + 0] = (excl + p0) * invT;
    s_cdf[wv][1 + lane * 4 + 1] = (excl + p1) * invT;
    s_cdf[wv][1 + lane * 4 + 2] = (excl + p2) * invT;
    s_cdf[wv][1 + lane * 4 + 3] = (excl + p3) * invT;
    wait_lds_0();   // order wave-private LDS stores before cross-lane gathers

    // ---- ray origin / direction (wave-uniform broadcast loads) ----
    const float* o3 = rayo + (size_t)ray * 3;
    const float* d3 = rayd + (size_t)ray * 3;
    const float ox = o3[0], oy = o3[1], oz = o3[2];
    const float dx = d3[0], dy = d3[1], dz = d3[2];

    // ---- 2 fine samples per lane: inverse-CDF + lerp + point ----
#pragma unroll
    for (int s = 0; s < 2; ++s) {
      const int   i = lane + s * 32;
      const float u = (float)i * (1.0f / (float)(NFS - 1));

      // right-insertion binary search over the 129-entry CDF
      int lo = 0, hi = SC + 1;
      while (lo < hi) {
        const int mid = (lo + hi) >> 1;
        if (s_cdf[wv][mid] <= u) lo = mid + 1; else hi = mid;
      }
      int below = lo - 1;
      if (below < 0)  below = 0;
      if (below > SC) below = SC;
      int above = lo;
      if (above > SC) above = SC;

      const float cb = s_cdf[wv][below];
      const float ca = s_cdf[wv][above];
      const int tb_i = (below > SC - 1) ? (SC - 1) : below;
      const int ta_i = (above > SC - 1) ? (SC - 1) : above;
      const float tb = s_t[wv][buf][tb_i];
      const float ta = s_t[wv][buf][ta_i];

      float denom = ca - cb;
      if (denom < 1e-5f) denom = 1.0f;
      const float frac = (u - cb) / denom;
      const float ft   = tb + frac * (ta - tb);

      const size_t ob = (size_t)ray * NFS + i;
      outt[ob] = ft;
      float* pp = outp + ob * 3;
      pp[0] = ox + dx * ft;
      pp[1] = oy + dy * ft;
      pp[2] = oz + dz * ft;
    }
  }
}

// ---- Host launch ---------------------------------------------------------

extern "C" void kernel_launch(void* const* d_in, const int* in_sizes, int n_in,
                              void* d_out, int out_size, void* d_ws, size_t ws_size,
                              hipStream_t stream) {
  (void)n_in; (void)d_ws; (void)ws_size; (void)out_size;
  const float* rayo = (const float*)d_in[0];   // [N,3]
  const float* rayd = (const float*)d_in[1];   // [N,3]
  const float* cw   = (const float*)d_in[2];   // [N,SC]
  const float* ct   = (const float*)d_in[3];   // [N,SC]
  const int n = in_sizes[0] / 3;

  float* outp = (float*)d_out;                  // fine_points [N,NFS,3]
  float* outt = outp + (size_t)n * NFS * 3;     // fine_t_vals [N,NFS]

  // 4096 blocks * 8 waves = 32768 waves -> 16 rays/wave at N=524288.
  int blocks = 4096;
  const int maxNeeded = (n + WAVES_PER_BLOCK - 1) / WAVES_PER_BLOCK;
  if (blocks > maxNeeded) blocks = maxNeeded;
  if (blocks < 1) blocks = 1;

  nerf_fine_sample_kernel<<<blocks, THREADS, 0, stream>>>(
      rayo, rayd, cw, ct, outp, outt, n);
}